// EdgeReconstructionLoss_558345749072
// MI455X (gfx1250) — compile-verified
//
#include <hip/hip_runtime.h>
#include <hip/hip_bf16.h>

typedef float v2f __attribute__((ext_vector_type(2)));
typedef float v8f __attribute__((ext_vector_type(8)));

__device__ __forceinline__ float softplus_f(float x) {
    // numerically stable: max(x,0) + log1p(exp(-|x|))
    return fmaxf(x, 0.0f) + log1pf(expf(-fabsf(x)));
}

// One wave processes a tile of 16 edges via V_WMMA_F32_16X16X4_F32 (fp32-exact).
// A = 16 gathered z_src rows, B = 16 gathered z_dst rows (as columns);
// diagonal of A*B = the 16 inner-product logits.
__global__ void edge_bce_wmma_kernel(const float* __restrict__ zsrc,
                                     const float* __restrict__ zdst,
                                     const int*   __restrict__ pidx,  // [2*E]
                                     const int*   __restrict__ nidx,  // [2*NEG]
                                     float*       __restrict__ partials,
                                     int E, int NEG)
{
    const int lane  = threadIdx.x & 31;
    const int wib   = threadIdx.x >> 5;          // wave in block
    const int wpb   = blockDim.x >> 5;           // waves per block
    const int gwave = blockIdx.x * wpb + wib;
    const int nwave = gridDim.x * wpb;

    const int posT = (E + 15) >> 4;
    const int negT = (NEG + 15) >> 4;
    const int totT = posT + negT;

    const int slot  = lane & 15;                 // edge slot within tile (M for A, N for B)
    const int khalf = (lane >> 4) << 1;          // lanes 16-31 hold K+2,K+3

    float acc = 0.0f;

    for (int t = gwave; t < totT; t += nwave) {  // t is wave-uniform -> EXEC stays full
        const int* __restrict__ idx;
        int base, n; float w, sgn;
        if (t < posT) { idx = pidx; base = t << 4;          n = E;   w = 0.5f / (float)E;   sgn = -1.0f; }
        else          { idx = nidx; base = (t - posT) << 4; n = NEG; w = 0.5f / (float)NEG; sgn =  1.0f; }

        int e  = base + slot;
        int ec = (e < n) ? e : (n - 1);          // clamp padded slots (weighted 0 below)
        const float* ap = zsrc + (size_t)idx[ec]     * 128 + khalf;  // A: M=slot, K=khalf..+1
        const float* bp = zdst + (size_t)idx[n + ec] * 128 + khalf;  // B: N=slot, K=khalf..+1

        v8f c = {0.f, 0.f, 0.f, 0.f, 0.f, 0.f, 0.f, 0.f};
        #pragma unroll 8
        for (int k = 0; k < 128; k += 4) {
            v2f a = *(const v2f*)(ap + k);
            v2f b = *(const v2f*)(bp + k);
            // (neg_a, A, neg_b, B, c_mod, C, reuse_a, reuse_b)
            c = __builtin_amdgcn_wmma_f32_16x16x4_f32(false, a, false, b,
                                                      (short)0, c, false, false);
        }

        // Diagonal extraction per C/D layout:
        //   (m,m) for m<8  -> lane m,    c[m]
        //   (m,m) for m>=8 -> lane m+16, c[m-8]
        int ci = (lane < 8) ? lane : (lane - 24);
        float s0 = (ci & 1) ? c[1] : c[0];
        float s1 = (ci & 1) ? c[3] : c[2];
        float s2 = (ci & 1) ? c[5] : c[4];
        float s3 = (ci & 1) ? c[7] : c[6];
        float u0 = (ci & 2) ? s1 : s0;
        float u1 = (ci & 2) ? s3 : s2;
        float logit = (ci & 4) ? u1 : u0;

        int  myedge = (lane < 8) ? (base + lane) : (base + lane - 16);
        bool have   = (lane < 8) || (lane >= 24);
        float contrib = (have && (myedge < n)) ? (w * softplus_f(sgn * logit)) : 0.0f;
        acc += contrib;
    }

    // wave32 reduction (fixed order -> deterministic)
    #pragma unroll
    for (int o = 16; o > 0; o >>= 1) acc += __shfl_down(acc, o, 32);

    __shared__ float smem[32];
    if (lane == 0) smem[wib] = acc;
    __syncthreads();
    if (wib == 0) {
        float s = (lane < wpb) ? smem[lane] : 0.0f;
        #pragma unroll
        for (int o = 16; o > 0; o >>= 1) s += __shfl_down(s, o, 32);
        if (lane == 0) partials[blockIdx.x] = s;
    }
}

__global__ void reduce_partials_kernel(const float* __restrict__ partials, int n,
                                       float* __restrict__ out)
{
    __shared__ float smem[256];
    float s = 0.0f;
    for (int i = threadIdx.x; i < n; i += blockDim.x) s += partials[i];  // fixed order
    smem[threadIdx.x] = s;
    __syncthreads();
    for (int o = 128; o > 0; o >>= 1) {
        if ((int)threadIdx.x < o) smem[threadIdx.x] += smem[threadIdx.x + o];
        __syncthreads();
    }
    if (threadIdx.x == 0) out[0] = smem[0];
}

extern "C" void kernel_launch(void* const* d_in, const int* in_sizes, int n_in,
                              void* d_out, int out_size, void* d_ws, size_t ws_size,
                              hipStream_t stream) {
    const float* zsrc = (const float*)d_in[0];
    const float* zdst = (const float*)d_in[1];
    const int*   pidx = (const int*)d_in[2];
    const int*   nidx = (const int*)d_in[3];
    const int E   = in_sizes[2] / 2;
    const int NEG = in_sizes[3] / 2;

    const int posT = (E + 15) >> 4;
    const int negT = (NEG + 15) >> 4;
    const int totT = posT + negT;

    const int threads = 256;                 // 8 wave32s per block
    const int wpb = threads / 32;
    int blocks = (totT + wpb - 1) / wpb;
    int cap = (int)(ws_size / sizeof(float));
    if (cap < 1) cap = 1;
    if (blocks > cap)  blocks = cap;
    if (blocks > 8192) blocks = 8192;        // grid-stride loop covers the rest
    if (blocks < 1)    blocks = 1;

    float* partials = (float*)d_ws;

    edge_bce_wmma_kernel<<<blocks, threads, 0, stream>>>(zsrc, zdst, pidx, nidx,
                                                         partials, E, NEG);
    reduce_partials_kernel<<<1, 256, 0, stream>>>(partials, blocks, (float*)d_out);
}